// WeightedAverage_1580547968506
// MI455X (gfx1250) — compile-verified
//
#include <hip/hip_runtime.h>

// ---------------------------------------------------------------------------
// WeightedAverage (3x3 local softmax recolorization stencil), MI455X / gfx1250
//
// Memory-bound stencil: ~42 MB of traffic -> ~1.8 us roofline at 23.3 TB/s.
// Strategy: 32x32 output tile per block, 34x34(+pad) halo of L/a/b staged in
// LDS via CDNA5 async global->LDS DMA (ASYNCcnt path), compute from LDS.
// ---------------------------------------------------------------------------

#define TILE     32
#define HALO     34           // TILE + 2 (radius-1 halo)
#define LDSW     36           // padded LDS row stride (floats) -> conflict-free
#define NTHREADS 256          // 8 waves (wave32)

typedef __attribute__((address_space(1))) int GINT;   // global int
typedef __attribute__((address_space(3))) int LINT;   // LDS int

// Async DMA: one 32-bit element per active lane, global -> LDS. ASYNCcnt.
__device__ __forceinline__ void async_copy_f32(const float* g, float* l) {
#if __has_builtin(__builtin_amdgcn_global_load_async_to_lds_b32)
    __builtin_amdgcn_global_load_async_to_lds_b32((GINT*)g, (LINT*)l, 0, 0);
#else
    unsigned l32 = (unsigned)(unsigned long long)
        (__attribute__((address_space(3))) float*)l;
    asm volatile("global_load_async_to_lds_b32 %0, %1, off"
                 :: "v"(l32), "v"(g) : "memory");
#endif
}

__device__ __forceinline__ void wait_async_zero() {
#if __has_builtin(__builtin_amdgcn_s_wait_asynccnt)
    __builtin_amdgcn_s_wait_asynccnt(0);
#else
    asm volatile("s_wait_asynccnt 0" ::: "memory");
#endif
}

__global__ __launch_bounds__(NTHREADS)
void wavg_softmax3x3(const float* __restrict__ x, float* __restrict__ out,
                     int H, int W)
{
    __shared__ float Lh[HALO * LDSW];
    __shared__ float Ah[HALO * LDSW];
    __shared__ float Bh[HALO * LDSW];

    const int tid = threadIdx.x;
    const int n   = blockIdx.z;
    const int x0  = blockIdx.x * TILE - 1;   // halo origin (global coords)
    const int y0  = blockIdx.y * TILE - 1;

    const size_t plane = (size_t)H * (size_t)W;
    const float* Lg = x + (size_t)n * 3 * plane;   // channel 0: L
    const float* Ag = Lg + plane;                  // channel 1: a
    const float* Bg = Lg + 2 * plane;              // channel 2: b

    // Boundary tiles: zero-fill LDS first so out-of-image neighbors behave
    // exactly like the reference's zero padding (they DO contribute
    // exp(-L_c^2) to the softmax denominator).
    const bool interior = (x0 >= 0) && (y0 >= 0) &&
                          (x0 + HALO <= W) && (y0 + HALO <= H);
    if (!interior) {
        for (int i = tid; i < HALO * LDSW; i += NTHREADS) {
            Lh[i] = 0.0f; Ah[i] = 0.0f; Bh[i] = 0.0f;
        }
        __syncthreads();   // ds stores complete before async DMA overwrites
    }

    // Stage the 34x34 halo of all three channels with async DMA.
    // Per-lane predication (EXEC mask) drops out-of-bounds transfers.
    for (int i = tid; i < HALO * HALO; i += NTHREADS) {
        const int ly = i / HALO;
        const int lx = i - ly * HALO;
        const int gy = y0 + ly;
        const int gx = x0 + lx;
        if (gy >= 0 && gy < H && gx >= 0 && gx < W) {
            const size_t go = (size_t)gy * W + gx;
            const int    lo = ly * LDSW + lx;
            async_copy_f32(Lg + go, &Lh[lo]);
            async_copy_f32(Ag + go, &Ah[lo]);
            async_copy_f32(Bg + go, &Bh[lo]);
        }
    }
    wait_async_zero();   // s_wait_asynccnt 0: DMA data landed in LDS
    __syncthreads();     // make it visible to all 8 waves

    // Compute: each thread produces 4 output pixels (rows ty0, ty0+8, ...).
    // Lanes 0..31 read consecutive LDS addresses -> bank-conflict free.
    const int tx  = tid & (TILE - 1);
    const int ty0 = tid >> 5;
    const int ox  = blockIdx.x * TILE + tx;

    float* outA = out + (size_t)n * 2 * plane;   // out channel 0
    float* outB = outA + plane;                  // out channel 1

#pragma unroll
    for (int k = 0; k < TILE / 8; ++k) {
        const int   ly = ty0 + k * 8;
        const int   c  = (ly + 1) * LDSW + (tx + 1);
        const float lc = Lh[c];

        float sw = 0.0f, sa = 0.0f, sb = 0.0f;
#pragma unroll
        for (int dy = 0; dy < 3; ++dy) {
#pragma unroll
            for (int dx = 0; dx < 3; ++dx) {
                const int   o = c + (dy - 1) * LDSW + (dx - 1);
                const float d = Lh[o] - lc;
                const float w = __expf(-d * d);   // v_exp_f32; w<=1, center w=1
                sw += w;
                sa = fmaf(w, Ah[o], sa);
                sb = fmaf(w, Bh[o], sb);
            }
        }
        // Softmax normalization (sw >= 1 always: center term is exp(0)).
        const float inv = __builtin_amdgcn_rcpf(sw);
        const size_t go = (size_t)(blockIdx.y * TILE + ly) * W + ox;
        outA[go] = sa * inv;
        outB[go] = sb * inv;
    }
}

extern "C" void kernel_launch(void* const* d_in, const int* in_sizes, int n_in,
                              void* d_out, int out_size, void* d_ws, size_t ws_size,
                              hipStream_t stream)
{
    (void)n_in; (void)out_size; (void)d_ws; (void)ws_size;
    const float* xin = (const float*)d_in[0];   // (N, 3, H, W) float32
    float*       out = (float*)d_out;           // (N, 2, H, W) float32

    const int H = 512, W = 512;
    const int N = in_sizes[0] / (3 * H * W);

    dim3 grid(W / TILE, H / TILE, N);           // (16, 16, N)
    wavg_softmax3x3<<<grid, NTHREADS, 0, stream>>>(xin, out, H, W);
}